// Script_DiaNet_43722767073564
// MI455X (gfx1250) — compile-verified
//
#include <hip/hip_runtime.h>
#include <math.h>

typedef __attribute__((ext_vector_type(2))) float v2f;
typedef __attribute__((ext_vector_type(8))) float v8f;

#define INP_NUM   20000
#define BATCH     1024
#define RED_HEAD  399
#define NLAYERS   398
#define NN_OUT    399
#define BT        16      // batch tile per block
#define NTHREADS  1024    // 32 waves: 1-2 row tiles per wave per layer
#define LDS_ROWS  800     // >= max dim 795 (+ scratch rows for OOB stores)
#define LDS_STR   17      // 16 batch + 1 pad float -> conflict-free LDS access
#define BUFSLOT   (LDS_ROWS * LDS_STR)

struct WPtrs { const float* w[NLAYERS]; };

__device__ __forceinline__ void layer_dims(int i, int& in_dim, int& out_dim, int& type) {
    if (i == 0)        { in_dim = 399;        out_dim = 397;        type = 0; }
    else if (i <= 199) { in_dim = 395 + 2*i;  out_dim = 397 + 2*i;  type = 1; }
    else               { in_dim = 1195 - 2*i; out_dim = 1193 - 2*i; type = 2; }
}

__device__ __forceinline__ float clip1(float w) {
#if __has_builtin(__builtin_amdgcn_fmed3f)
    return __builtin_amdgcn_fmed3f(w, -1.0f, 1.0f);   // single v_med3
#else
    return fminf(1.0f, fmaxf(-1.0f, w));
#endif
}

__device__ __forceinline__ float fast_tanh(float v) {
#if __has_builtin(__builtin_amdgcn_tanhf)
    return __builtin_amdgcn_tanhf(v);
#else
    return tanhf(v);
#endif
}

extern __shared__ float lds[];

__global__ __launch_bounds__(NTHREADS)
void dianet_kernel(const float* __restrict__ x, WPtrs wp, float* __restrict__ out) {
    const int tid    = threadIdx.x;
    const int lane   = tid & 31;
    const int wave   = tid >> 5;
    const int nwaves = blockDim.x >> 5;
    const int lg     = lane >> 4;   // lane group (0/1): selects K pair for A/B frags
    const int ln     = lane & 15;   // 0..15: A row (M) / B-D column (batch N)
    const int batch0 = blockIdx.x * BT;

    // stage layer-0 input x[:, 0:399] into buffer slot 2 ("output of layer -1")
    {
        float* b2 = lds + 2 * BUFSLOT;
        for (int idx = tid; idx < RED_HEAD * BT; idx += blockDim.x) {
            int r = idx / BT, b = idx % BT;
            b2[r * LDS_STR + b] = x[(size_t)(batch0 + b) * INP_NUM + r];
        }
    }
    __syncthreads();

    const float* __restrict__ xrow = x + (size_t)(batch0 + ln) * INP_NUM;

    for (int layer = 0; layer < NLAYERS; ++layer) {
        int in_dim, out_dim, type;
        layer_dims(layer, in_dim, out_dim, type);
        const int nm1 = in_dim - 1;
        const float* __restrict__ W = wp.w[layer];
        float* bin  = lds + ((layer + 2) % 3) * BUFSLOT;   // output of layer-1
        float* bh2  = lds + ((layer + 1) % 3) * BUFSLOT;   // output of layer-2 (skip)
        float* bout = lds + (layer % 3) * BUFSLOT;
        int hid2_dim = 0;
        if (layer >= 2) { int a, b, t; layer_dims(layer - 2, a, b, t); hid2_dim = b; }
        const int dhalf = (layer >= 2) ? (out_dim - hid2_dim) / 2 : 0;  // +2/0/-2

        // input-injection parameters (layers 1..198); folded into B at odd columns
        const bool inj    = (layer >= 1 && layer <= 198);
        const int sel_len = 199 - layer;
        const int start_j = ((197 + layer) >> 1) - (sel_len >> 1);
        const int seg_off = RED_HEAD + (layer - 1) * 199 - ((layer - 1) * layer) / 2;

        const int nrt = (out_dim + 15) >> 4;
        for (int rt = wave; rt < nrt; rt += nwaves) {
            const int r0   = rt << 4;
            const int arow = r0 + ln;                    // A row held by this lane
            const bool row_ok = (arow < out_dim) && !(arow & 1);
            const float* __restrict__ wrow = W + (size_t)min(arow, out_dim - 1) * in_dim;
            const int klo   = (r0 == 0) ? 0 : (r0 - 4);  // 4-aligned; band = 6 chunks
            const int kbase = klo + 2 * lg;              // lane's first (even) column

            // ---- band bitmask relative to klo (bits 0..23), per-lane row ----
            unsigned am;
            {
                const int r = arow;
                int i0, i1, i2, i3, i4;
                if (type == 0)      { i0 = r;              i1 = r + 1;     i2 = r + 2;
                                      i3 = r;              i4 = r; }
                else if (type == 1) { i0 = max(0, r - 4);  i1 = min(max(0, r - 2), nm1);
                                      i2 = min(max(0, r - 1), nm1);
                                      i3 = min(r, nm1);    i4 = min(r + 2, nm1); }
                else                { i0 = max(0, r - 2);  i1 = min(r, nm1);
                                      i2 = min(r + 1, nm1);
                                      i3 = min(r + 2, nm1); i4 = min(r + 4, nm1); }
                unsigned bandm = (1u << (i0 - klo)) | (1u << (i1 - klo))
                               | (1u << (i2 - klo)) | (1u << (i3 - klo))
                               | (1u << (i4 - klo));
                am = (row_ok ? bandm : 0u) >> (2 * lg);  // chunk ci -> bits 4ci, 4ci+1
            }
            // valid-column prefix mask (cols klo..nm1), uniform per tile
            const int vcnt = nm1 - klo + 1;
            const unsigned vfull = (vcnt >= 32) ? 0xFFFFFFFFu
                                 : ((vcnt <= 0) ? 0u : ((1u << vcnt) - 1u));
            const unsigned vm = vfull >> (2 * lg);

            // ---- phase 1: issue ALL loads (12 global + 12 LDS), batched ----
            float wv[12], sv[12];
            #pragma unroll
            for (int ci = 0; ci < 6; ++ci) {
                const int ka = kbase + 4 * ci;           // even
                const int kb = ka + 1;                   // odd
                wv[2*ci]   = wrow[min(ka, nm1)];
                wv[2*ci+1] = wrow[min(kb, nm1)];
            }
            #pragma unroll
            for (int ci = 0; ci < 6; ++ci) {
                const int ka = kbase + 4 * ci;
                const int kb = ka + 1;
                sv[2*ci]   = bin[min(ka, nm1) * LDS_STR + ln];
                sv[2*ci+1] = bin[min(kb, nm1) * LDS_STR + ln];
            }
            // injection: add seg value to B at odd columns (only layers 1..198)
            if (inj) {
                #pragma unroll
                for (int ci = 0; ci < 6; ++ci) {
                    const int kb  = kbase + 4 * ci + 1;  // odd column = 2j+1
                    const int j   = kb >> 1;
                    const int col = seg_off + (j - start_j);
                    const bool ok = (j >= start_j) && (j < start_j + sel_len)
                                 && (col < INP_NUM);
                    const float xv = xrow[min(col, INP_NUM - 1)];
                    sv[2*ci+1] += ok ? xv : 0.0f;
                }
            }

            // ---- phase 2+3: build fragments, 6 WMMAs (dual accumulators) ----
            v8f c0 = {0.f,0.f,0.f,0.f,0.f,0.f,0.f,0.f};
            v8f c1 = {0.f,0.f,0.f,0.f,0.f,0.f,0.f,0.f};
            #pragma unroll
            for (int ci = 0; ci < 6; ++ci) {
                const bool h0 = (am >> (4 * ci))     & 1u;
                const bool h1 = (am >> (4 * ci + 1)) & 1u;
                const bool g0 = (vm >> (4 * ci))     & 1u;
                const bool g1 = (vm >> (4 * ci + 1)) & 1u;
                v2f a, b;
                a.x = h0 ? clip1(wv[2*ci])   : 0.0f;
                a.y = h1 ? clip1(wv[2*ci+1]) : 0.0f;
                b.x = g0 ? sv[2*ci]   : 0.0f;
                b.y = g1 ? sv[2*ci+1] : 0.0f;
                if (ci & 1)
                    c1 = __builtin_amdgcn_wmma_f32_16x16x4_f32(false, a, false, b,
                                                               (short)0, c1, false, false);
                else
                    c0 = __builtin_amdgcn_wmma_f32_16x16x4_f32(false, a, false, b,
                                                               (short)0, c0, false, false);
            }

            // ---- epilogue: skip add + tanh + store, fully branchless ----
            #pragma unroll
            for (int v = 0; v < 8; ++v) {
                const int r = r0 + v + 8 * lg;           // D row: VGPR v, lane group lg
                float acc = c0[v] + c1[v];
                const int src  = r - dhalf;
                const int srcc = min(max(src, 0), LDS_ROWS - 1);
                const float h2 = bh2[srcc * LDS_STR + ln];
                const bool okh = (layer >= 2) && (src >= 0) && (src < hid2_dim);
                acc += okh ? h2 : 0.0f;
                // rows >= out_dim land in never-read LDS rows (< LDS_ROWS): safe
                bout[r * LDS_STR + ln] = fast_tanh(acc);
            }
        }
        __syncthreads();
    }

    // result = output of layer 397 -> slot (398-1) % 3 == 1; write [BATCH, 399]
    const float* bres = lds + ((NLAYERS - 1) % 3) * BUFSLOT;
    for (int idx = tid; idx < NN_OUT * BT; idx += blockDim.x) {
        int r = idx / BT, b = idx % BT;
        out[(size_t)(batch0 + b) * NN_OUT + r] = bres[r * LDS_STR + b];
    }
}

extern "C" void kernel_launch(void* const* d_in, const int* in_sizes, int n_in,
                              void* d_out, int out_size, void* d_ws, size_t ws_size,
                              hipStream_t stream) {
    (void)in_sizes; (void)n_in; (void)out_size; (void)d_ws; (void)ws_size;
    const float* x = (const float*)d_in[0];
    WPtrs wp;
    for (int i = 0; i < NLAYERS; ++i) wp.w[i] = (const float*)d_in[1 + i];

    const size_t shmem = (size_t)3 * BUFSLOT * sizeof(float); // 163200 B
    (void)hipFuncSetAttribute((const void*)dianet_kernel,
                              hipFuncAttributeMaxDynamicSharedMemorySize, (int)shmem);
    dianet_kernel<<<BATCH / BT, NTHREADS, shmem, stream>>>(x, wp, (float*)d_out);
}